// DinkNet_19026705121763
// MI455X (gfx1250) — compile-verified
//
#include <hip/hip_runtime.h>
#include <hip/hip_bf16.h>
#include <stdint.h>

typedef __attribute__((ext_vector_type(2))) float v2f;
typedef __attribute__((ext_vector_type(8))) float v8f;

#define NIN 256
#define NH  128
#define AST 260   // LDS row stride in floats: %64 == 4 -> conflict-free, 16B-aligned rows

// ---------------------------------------------------------------------------
// Precompute wsum[k] = sum_j lin_W[k][j], bsum = sum(lin_b). One block, 128 thr.
// ---------------------------------------------------------------------------
__global__ void wsum_kernel(const float* __restrict__ linW,
                            const float* __restrict__ linb,
                            float* __restrict__ wsum, float* __restrict__ bsum) {
    int t = threadIdx.x;
    float s = 0.f;
    #pragma unroll 4
    for (int j = 0; j < NH; ++j) s += linW[t * NH + j];
    wsum[t] = s;
    if (t == 0) {
        float b = 0.f;
        for (int j = 0; j < NH; ++j) b += linb[j];
        *bsum = b;
    }
}

// ---------------------------------------------------------------------------
// H = X @ W   (X: [n,256] row-major, W: [256,128] row-major, H: [n,128])
// Block: 256 threads = 8 waves. Block owns a 16-row stripe; wave w owns
// columns [16w, 16w+16). A-tile staged in LDS via GLOBAL_LOAD_ASYNC_TO_LDS;
// 64x V_WMMA_F32_16X16X4_F32 per wave.
// ---------------------------------------------------------------------------
__global__ void gemm_kernel(const float* __restrict__ X,
                            const float* __restrict__ W,
                            float* __restrict__ H, int n) {
    __shared__ __align__(16) float Alds[16 * AST];

    const int tile = blockIdx.x;
    const int t    = threadIdx.x;
    const size_t base = (size_t)tile * 16 * NIN;
    const bool full_tile = (tile * 16 + 16) <= n;

    // Stage the 16x256 A tile in LDS: 1024 float4, 4 per thread.
    // Fast path: async global->LDS DMA (no VGPR round-trip, ASYNCcnt-tracked).
    #pragma unroll
    for (int i = 0; i < 4; ++i) {
        int g    = t + i * 256;     // float4 index within the tile
        int row  = g >> 6;          // 64 float4 per row
        int col4 = g & 63;
        float*   lp = &Alds[row * AST + col4 * 4];
        if (full_tile || (tile * 16 + row < n)) {
            uint32_t la = (uint32_t)(uintptr_t)lp;          // LDS byte offset
            uint64_t ga = (uint64_t)(uintptr_t)(X + base + (size_t)g * 4);
            asm volatile("global_load_async_to_lds_b128 %0, %1, off"
                         :: "v"(la), "v"(ga) : "memory");
        } else {
            *(float4*)lp = make_float4(0.f, 0.f, 0.f, 0.f);
        }
    }
    asm volatile("s_wait_asynccnt 0x0" ::: "memory");
    __syncthreads();

    const int wave = t >> 5;
    const int lane = t & 31;
    const int lhi  = lane >> 4;     // 0: lanes 0-15, 1: lanes 16-31
    const int lm   = lane & 15;
    const int n0   = wave * 16;

    const float* __restrict__ arow = &Alds[lm * AST];
    const float* __restrict__ wcol = W + n0 + lm;

    v8f c = {};
    #pragma unroll 8
    for (int k = 0; k < NIN; k += 4) {
        const int kk = k + lhi * 2;                 // this half-wave's K pair
        // A fragment: lane (lm, half lhi) holds A[lm][kk], A[lm][kk+1]
        float2 av = *(const float2*)(arow + kk);    // kk even -> 8B aligned
        v2f a = {av.x, av.y};
        // B fragment: lane holds B[kk][n0+lm], B[kk+1][n0+lm]
        v2f b = {wcol[kk * NH], wcol[(kk + 1) * NH]};
        c = __builtin_amdgcn_wmma_f32_16x16x4_f32(
                /*neg_a=*/false, a, /*neg_b=*/false, b,
                /*c_mod=*/(short)0, c, /*reuse_a=*/false, /*reuse_b=*/false);
    }

    // C/D layout: VGPR r -> M = r (lanes 0-15) / r+8 (lanes 16-31), N = lane&15.
    float* __restrict__ outp = H + ((size_t)tile * 16 + lhi * 8) * NH + n0 + lm;
    if (full_tile) {
        #pragma unroll
        for (int r = 0; r < 8; ++r) outp[(size_t)r * NH] = c[r];
    } else {
        #pragma unroll
        for (int r = 0; r < 8; ++r)
            if (tile * 16 + lhi * 8 + r < n) outp[(size_t)r * NH] = c[r];
    }
}

// ---------------------------------------------------------------------------
// Zero AGG.
// ---------------------------------------------------------------------------
__global__ void zero_kernel(float4* __restrict__ p, int n4) {
    int i = blockIdx.x * blockDim.x + threadIdx.x;
    if (i < n4) p[i] = make_float4(0.f, 0.f, 0.f, 0.f);
}

// ---------------------------------------------------------------------------
// AGG[row_e][:] += w_e * H[col_e][:]. 128 threads per edge, 2 edges per block.
// AGG and H both fit in the 192MB L2 -> atomics and gathers stay on-chip.
// ---------------------------------------------------------------------------
__global__ void scatter_kernel(const int* __restrict__ rows,
                               const int* __restrict__ cols,
                               const float* __restrict__ ew,
                               const float* __restrict__ H,
                               float* __restrict__ AGG, int E) {
    int e = blockIdx.x * 2 + (threadIdx.x >> 7);
    int t = threadIdx.x & 127;
    if (e >= E) return;
    int   r = rows[e];
    int   cidx = cols[e];
    float w = ew[e];
    float v = w * H[(size_t)cidx * NH + t];
    unsafeAtomicAdd(&AGG[(size_t)r * NH + t], v);
}

// ---------------------------------------------------------------------------
// z[i] = dot(prelu(AGG[i] + bias), wsum) + bsum. One wave32 per node,
// float4 per lane, shuffle reduction.
// ---------------------------------------------------------------------------
__global__ void finalize_kernel(const float* __restrict__ AGG,
                                const float* __restrict__ bias,
                                const float* __restrict__ prelu_a,
                                const float* __restrict__ wsum,
                                const float* __restrict__ bsum,
                                float* __restrict__ z, int n) {
    int wave = threadIdx.x >> 5;
    int lane = threadIdx.x & 31;
    int node = blockIdx.x * 8 + wave;
    if (node >= n) return;

    float a = prelu_a[0];
    float4 v  = *(const float4*)(AGG + (size_t)node * NH + lane * 4);
    float4 b4 = *(const float4*)(bias + lane * 4);
    float4 w4 = *(const float4*)(wsum + lane * 4);

    float x0 = v.x + b4.x; x0 = (x0 >= 0.f) ? x0 : a * x0;
    float x1 = v.y + b4.y; x1 = (x1 >= 0.f) ? x1 : a * x1;
    float x2 = v.z + b4.z; x2 = (x2 >= 0.f) ? x2 : a * x2;
    float x3 = v.w + b4.w; x3 = (x3 >= 0.f) ? x3 : a * x3;
    float s = x0 * w4.x + x1 * w4.y + x2 * w4.z + x3 * w4.w;

    #pragma unroll
    for (int off = 16; off > 0; off >>= 1) s += __shfl_xor(s, off, 32);

    if (lane == 0) z[node] = s + bsum[0];
}

// ---------------------------------------------------------------------------
extern "C" void kernel_launch(void* const* d_in, const int* in_sizes, int n_in,
                              void* d_out, int out_size, void* d_ws, size_t ws_size,
                              hipStream_t stream) {
    const float* x1 = (const float*)d_in[0];
    const float* x2 = (const float*)d_in[1];
    const int*   ei = (const int*)d_in[2];
    const float* ew = (const float*)d_in[3];
    const float* Wg = (const float*)d_in[4];
    const float* bg = (const float*)d_in[5];
    const float* pa = (const float*)d_in[6];
    const float* lW = (const float*)d_in[7];
    const float* lb = (const float*)d_in[8];
    float* out = (float*)d_out;

    const int N = in_sizes[0] / NIN;       // 100000
    const int E = in_sizes[2] / 2;         // 800000

    float* ws   = (float*)d_ws;
    float* H    = ws;                      // [N,128]
    float* AGG  = ws + (size_t)N * NH;     // [N,128]
    float* wsum = ws + (size_t)2 * N * NH; // [128]
    float* bsum = wsum + NH;               // [1]

    wsum_kernel<<<1, NH, 0, stream>>>(lW, lb, wsum, bsum);

    const float* xs[2] = {x1, x2};
    const int ntiles = (N + 15) / 16;
    const int n4     = (N * NH) / 4;

    for (int j = 0; j < 2; ++j) {
        gemm_kernel<<<ntiles, 256, 0, stream>>>(xs[j], Wg, H, N);
        zero_kernel<<<(n4 + 255) / 256, 256, 0, stream>>>((float4*)AGG, n4);
        scatter_kernel<<<(E + 1) / 2, 256, 0, stream>>>(ei, ei + E, ew, H, AGG, E);
        finalize_kernel<<<(N + 7) / 8, 256, 0, stream>>>(AGG, bg, pa, wsum, bsum,
                                                         out + (size_t)j * N, N);
    }
}